// NMSLayer_45578192945505
// MI455X (gfx1250) — compile-verified
//
#include <hip/hip_runtime.h>

// ---------------------------------------------------------------------------
// NMS-like op: global median -> threshold -> 7x7 maxpool -> keep local maxima.
// MI455X plan: memory-bound (134MB in / 134MB out, ~0.5 G fmax ops -> AI ~ 1);
// 3-pass radix select (tensor fits in 192MB L2 after pass 1) + fused LDS-tiled
// pool whose interior tiles are staged via gfx1250 async-to-LDS DMA (ASYNCcnt).
// ---------------------------------------------------------------------------

#define H0_OFF     0
#define H1_OFF     2048
#define H2_OFF     4096
#define WS_RANK    5120
#define WS_PREFIX  5121
#define WS_MEDBITS 5122
#define WS_WORDS   5123

#define TILE   64
#define HALO   3
#define LW     70   // TILE + 2*HALO (window rows/cols)
#define LPITCH 72   // staged cols: 70 window cols + 1 pad each side -> 18 float4

// Order-preserving float -> uint key (total order; no NaNs in this workload).
__device__ __forceinline__ unsigned fkey(float f) {
  unsigned u = __float_as_uint(f);
  return (u & 0x80000000u) ? ~u : (u | 0x80000000u);
}
__device__ __forceinline__ float kinv(unsigned k) {
  unsigned u = (k & 0x80000000u) ? (k ^ 0x80000000u) : ~k;
  return __uint_as_float(u);
}

// ---------------------------------------------------------------------------
// Workspace init: zero histograms, seed rank=k0, prefix=0. Runs every call
// (harness poisons ws once and never re-poisons; we must self-initialize).
// ---------------------------------------------------------------------------
__global__ void ws_init_kernel(unsigned* __restrict__ ws, unsigned k0) {
  int i = blockIdx.x * blockDim.x + threadIdx.x;
  if (i < WS_RANK) ws[i] = 0u;
  else if (i == WS_RANK) ws[i] = k0;
  else if (i == WS_PREFIX) ws[i] = 0u;
  else if (i == WS_MEDBITS) ws[i] = 0u;
}

// ---------------------------------------------------------------------------
// Radix-select histogram pass. Pass1: bin = key>>21 (no filter).
// Pass2: filter key>>21 == prefix11, bin = (key>>10)&0x7FF.
// Pass3: filter key>>10 == prefix22, bin = key & 0x3FF.
// LDS-private 2048-bin histogram (ds_add_u32), sparse flush via global atomics.
// float4 loads -> global_load_b128; software prefetch -> global_prefetch_b8.
// ---------------------------------------------------------------------------
__global__ void hist_kernel(const float4* __restrict__ x4, long long n4,
                            unsigned* __restrict__ ws, int histOff,
                            int filterShift, int binShift, unsigned binMask,
                            int useFilter) {
  __shared__ unsigned lh[2048];
  for (int i = threadIdx.x; i < 2048; i += blockDim.x) lh[i] = 0u;
  __syncthreads();

  const unsigned prefix = useFilter ? ws[WS_PREFIX] : 0u;  // uniform -> s_load
  const long long stride = (long long)gridDim.x * blockDim.x;

  for (long long i = (long long)blockIdx.x * blockDim.x + threadIdx.x;
       i < n4; i += stride) {
    __builtin_prefetch(&x4[i + stride], 0, 0);  // speculative, dropped if OOB
    float4 v = x4[i];
    unsigned k0 = fkey(v.x), k1 = fkey(v.y), k2 = fkey(v.z), k3 = fkey(v.w);
    if (!useFilter) {
      atomicAdd(&lh[(k0 >> binShift) & binMask], 1u);
      atomicAdd(&lh[(k1 >> binShift) & binMask], 1u);
      atomicAdd(&lh[(k2 >> binShift) & binMask], 1u);
      atomicAdd(&lh[(k3 >> binShift) & binMask], 1u);
    } else {
      if ((k0 >> filterShift) == prefix) atomicAdd(&lh[(k0 >> binShift) & binMask], 1u);
      if ((k1 >> filterShift) == prefix) atomicAdd(&lh[(k1 >> binShift) & binMask], 1u);
      if ((k2 >> filterShift) == prefix) atomicAdd(&lh[(k2 >> binShift) & binMask], 1u);
      if ((k3 >> filterShift) == prefix) atomicAdd(&lh[(k3 >> binShift) & binMask], 1u);
    }
  }
  __syncthreads();
  for (int i = threadIdx.x; i < 2048; i += blockDim.x) {
    unsigned c = lh[i];
    if (c) atomicAdd(&ws[histOff + i], c);  // sparse flush: ~free on passes 2/3
  }
}

// ---------------------------------------------------------------------------
// Single-workgroup (1024 threads) bin selection: two-level prefix scan over
// <=2048 bins, find bin containing current rank, narrow rank + extend prefix.
// On the final pass, reconstruct the median float and stash its bits.
// ---------------------------------------------------------------------------
__global__ void select_kernel(unsigned* __restrict__ ws, int histOff, int nbins,
                              int bits, int finalPass) {
  __shared__ unsigned s[2048];
  __shared__ unsigned t2[1024];
  const int tid = threadIdx.x;

  // Read rank BEFORE any thread can overwrite it (barriers below order this).
  const unsigned rank = ws[WS_RANK];

  s[tid]        = (tid < nbins)        ? ws[histOff + tid]        : 0u;
  s[tid + 1024] = (tid + 1024 < nbins) ? ws[histOff + tid + 1024] : 0u;
  __syncthreads();

  const unsigned a = s[2 * tid], b = s[2 * tid + 1];
  const unsigned c = a + b;
  t2[tid] = c;
  __syncthreads();
  for (int off = 1; off < 1024; off <<= 1) {       // Hillis-Steele over chunk sums
    unsigned v = (tid >= off) ? t2[tid - off] : 0u;
    __syncthreads();
    t2[tid] += v;
    __syncthreads();
  }
  const unsigned base = t2[tid] - c;     // exclusive prefix of this thread's 2 bins
  const unsigned cum0 = base + a;
  const unsigned cum1 = cum0 + b;

  int bin = -1;
  unsigned newRank = 0u;
  if (rank >= base && rank < cum0)      { bin = 2 * tid;     newRank = rank - base; }
  else if (rank >= cum0 && rank < cum1) { bin = 2 * tid + 1; newRank = rank - cum0; }

  if (bin >= 0) {  // exactly one winner (total count > rank by construction)
    unsigned p = (ws[WS_PREFIX] << bits) | (unsigned)bin;
    ws[WS_PREFIX] = p;
    ws[WS_RANK]   = newRank;
    if (finalPass) ws[WS_MEDBITS] = __float_as_uint(kinv(p));
  }
}

// ---------------------------------------------------------------------------
// Fused threshold + separable 7x7 maxpool + local-max select.
// 64x64 output tile, halo 3. Interior tiles (~88% of blocks): stage a 70x72
// region (16B-aligned: first col = bx*64-4) straight into LDS with
// global_load_async_to_lds_b128 (ASYNCcnt path, no VGPR staging, no bounds
// checks), wait with s_wait_asynccnt, then threshold in place. Boundary tiles:
// scalar loads with exact -inf OOB padding (reference pads -inf).
// Window col c lives at LDS col c+1. ~38KB LDS -> 8 workgroups per 320KB WGP.
// ---------------------------------------------------------------------------
__global__ void nms_kernel(const float* __restrict__ x,
                           const unsigned* __restrict__ ws,
                           float* __restrict__ out, int H, int W) {
  __shared__ __align__(16) float s_thr[LW * LPITCH];   // 70 x 72 staged floats
  __shared__ float s_rmax[LW * TILE];

  const float med = __uint_as_float(ws[WS_MEDBITS]);
  const int oy  = blockIdx.y * TILE - HALO;       // first window row
  const int ox2 = blockIdx.x * TILE - HALO - 1;   // first STAGED col (16B aligned)
  const long long plane = (long long)H * W;
  const float* xp = x + (long long)blockIdx.z * plane;
  float* op = out + (long long)blockIdx.z * plane;
  const float NEG_INF = -__builtin_huge_valf();

  // Block-uniform predicate -> in-branch barriers are safe.
  const bool interior =
      (ox2 >= 0) && (ox2 + LPITCH <= W) && (oy >= 0) && (oy + LW <= H);

  if (interior) {
    // Low 32 bits of a generic LDS pointer == LDS byte address (ISA 10.2).
    const unsigned lds_base = (unsigned)(unsigned long long)(const void*)s_thr;
    // 70 rows x 18 float4 = 1260 async B128 transfers per block (~5/thread).
    for (int i = threadIdx.x; i < LW * (LPITCH / 4); i += blockDim.x) {
      int r = i / (LPITCH / 4);
      int q = i - r * (LPITCH / 4);
      unsigned long long gaddr = (unsigned long long)(const void*)(
          xp + (long long)(oy + r) * W + (ox2 + 4 * q));
      unsigned laddr = lds_base + (unsigned)(i * 16);
      asm volatile("global_load_async_to_lds_b128 %0, %1, off"
                   :: "v"(laddr), "v"(gaddr)
                   : "memory");
    }
    asm volatile("s_wait_asynccnt 0x0" ::: "memory");
    __syncthreads();
    // Threshold in place (every staged element is in-bounds for interior tiles).
    for (int i = threadIdx.x; i < LW * LPITCH; i += blockDim.x) {
      float t = s_thr[i];
      s_thr[i] = (t > med) ? t : 0.0f;
    }
  } else {
    // Boundary tiles: fill the 70 window cols (LDS cols 1..70); -inf for OOB.
    for (int i = threadIdx.x; i < LW * LW; i += blockDim.x) {
      int r = i / LW, c2 = i - r * LW;
      int gy = oy + r, gx = ox2 + 1 + c2;
      float v = NEG_INF;
      if (gy >= 0 && gy < H && gx >= 0 && gx < W) {
        float t = xp[(long long)gy * W + gx];
        v = (t > med) ? t : 0.0f;
      }
      s_thr[r * LPITCH + (c2 + 1)] = v;
    }
  }
  __syncthreads();

  // Horizontal 7-wide max for all 70 rows x 64 output cols (window col +1).
  for (int i = threadIdx.x; i < LW * TILE; i += blockDim.x) {
    int r = i / TILE, c2 = i - r * TILE;
    const float* row = &s_thr[r * LPITCH + c2 + 1];
    float m = row[0];
    m = fmaxf(m, row[1]); m = fmaxf(m, row[2]); m = fmaxf(m, row[3]);
    m = fmaxf(m, row[4]); m = fmaxf(m, row[5]); m = fmaxf(m, row[6]);
    s_rmax[r * TILE + c2] = m;
  }
  __syncthreads();

  // Vertical 7-wide max + local-max select + store.
  for (int i = threadIdx.x; i < TILE * TILE; i += blockDim.x) {
    int r = i / TILE, c2 = i - r * TILE;
    const float* col = &s_rmax[r * TILE + c2];
    float m = col[0];
    m = fmaxf(m, col[1 * TILE]); m = fmaxf(m, col[2 * TILE]);
    m = fmaxf(m, col[3 * TILE]); m = fmaxf(m, col[4 * TILE]);
    m = fmaxf(m, col[5 * TILE]); m = fmaxf(m, col[6 * TILE]);
    float center = s_thr[(r + HALO) * LPITCH + (c2 + HALO) + 1];
    int gy = blockIdx.y * TILE + r, gx = blockIdx.x * TILE + c2;
    if (gy < H && gx < W)
      op[(long long)gy * W + gx] = (center == m) ? center : 0.0f;
  }
}

// ---------------------------------------------------------------------------
extern "C" void kernel_launch(void* const* d_in, const int* in_sizes, int n_in,
                              void* d_out, int out_size, void* d_ws, size_t ws_size,
                              hipStream_t stream) {
  (void)n_in; (void)out_size; (void)ws_size;
  const float* x = (const float*)d_in[0];
  float* out = (float*)d_out;
  unsigned* ws = (unsigned*)d_ws;

  const long long n = (long long)in_sizes[0];          // 8*1*2048*2048
  const unsigned k0 = (unsigned)((n - 1) / 2);         // lower-median rank
  const long long n4 = n / 4;                          // float4 count
  const int H = 2048, W = 2048;
  const int B = (int)(n / ((long long)H * W));

  ws_init_kernel<<<(WS_WORDS + 1023) / 1024, 1024, 0, stream>>>(ws, k0);

  // Pass 1: top 11 bits.
  hist_kernel<<<2048, 256, 0, stream>>>((const float4*)x, n4, ws, H0_OFF,
                                        0, 21, 0x7FFu, 0);
  select_kernel<<<1, 1024, 0, stream>>>(ws, H0_OFF, 2048, 11, 0);
  // Pass 2: next 11 bits (reads should hit the 192MB L2).
  hist_kernel<<<2048, 256, 0, stream>>>((const float4*)x, n4, ws, H1_OFF,
                                        21, 10, 0x7FFu, 1);
  select_kernel<<<1, 1024, 0, stream>>>(ws, H1_OFF, 2048, 11, 0);
  // Pass 3: final 10 bits -> exact median bit pattern.
  hist_kernel<<<2048, 256, 0, stream>>>((const float4*)x, n4, ws, H2_OFF,
                                        10, 0, 0x3FFu, 1);
  select_kernel<<<1, 1024, 0, stream>>>(ws, H2_OFF, 1024, 10, 1);

  // Fused threshold + 7x7 maxpool + local-max select.
  dim3 grid(W / TILE, H / TILE, B);
  nms_kernel<<<grid, 256, 0, stream>>>(x, ws, out, H, W);
}